// ListwiseSmoothINDCGKLoss_57681410785436
// MI455X (gfx1250) — compile-verified
//
#include <hip/hip_runtime.h>
#include <math.h>

#define ALPHA_C 10.0f
#define DELTA_C 0.1f
#define KTOP 10
#define EPS_C 1e-10f
#define LL 2048
#define TPB 256
#define EPT 8  // elements per thread: 2048 / 256
#define LOG2E_C 1.4426950408889634f

// ---- fast hardware transcendentals (guarded) ----
#if __has_builtin(__builtin_amdgcn_exp2f)
__device__ __forceinline__ float fast_exp2(float x) { return __builtin_amdgcn_exp2f(x); }
#else
__device__ __forceinline__ float fast_exp2(float x) { return exp2f(x); }
#endif

#if __has_builtin(__builtin_amdgcn_logf)
__device__ __forceinline__ float fast_log2(float x) { return __builtin_amdgcn_logf(x); }
#else
__device__ __forceinline__ float fast_log2(float x) { return log2f(x); }
#endif

#if __has_builtin(__builtin_amdgcn_rcpf)
__device__ __forceinline__ float fast_rcp(float x) { return __builtin_amdgcn_rcpf(x); }
#else
__device__ __forceinline__ float fast_rcp(float x) { return 1.0f / x; }
#endif

// ---- CDNA5 async global->LDS staging (guarded) ----
// Builtin signature (from hipcc diagnostic): arg0 = int4* in AS(1) (global),
// arg1 = LDS pointer, then imm offset + imm cpol.
typedef int v4i __attribute__((vector_size(16)));
typedef __attribute__((address_space(1))) v4i* gv4i_p;
typedef __attribute__((address_space(3))) v4i* lv4i_p;

#if __has_builtin(__builtin_amdgcn_global_load_async_to_lds_b128)
#define HAVE_ASYNC_LDS 1
#endif

__device__ __forceinline__ void wait_async_lds() {
#if defined(HAVE_ASYNC_LDS)
#if __has_builtin(__builtin_amdgcn_s_wait_asynccnt)
    __builtin_amdgcn_s_wait_asynccnt(0);
#else
    asm volatile("s_wait_asynccnt 0" ::: "memory");
#endif
#endif
}

__global__ __launch_bounds__(TPB) void ndcg_rows_kernel(const float* __restrict__ s_g,
                                                        const float* __restrict__ l_g,
                                                        float* __restrict__ row_out) {
    __shared__ __align__(16) float s_sh[LL];
    __shared__ __align__(16) float l_sh[LL];
    __shared__ float red[8];      // one partial per wave (8 waves)
    __shared__ float2 red2[8];
    __shared__ int hist[8];       // 5 used

    const int row  = blockIdx.x;
    const int t    = threadIdx.x;
    const int base = t * EPT;
    const int lane = t & 31;      // wave32
    const int wid  = t >> 5;

    const float* gs = s_g + (size_t)row * LL + base;
    const float* gl = l_g + (size_t)row * LL + base;

#if defined(HAVE_ASYNC_LDS)
    // Stage this row into LDS via the CDNA5 async data path (tracked by ASYNCcnt).
    __builtin_amdgcn_global_load_async_to_lds_b128((gv4i_p)(gs),     (lv4i_p)(&s_sh[base]),     0, 0);
    __builtin_amdgcn_global_load_async_to_lds_b128((gv4i_p)(gs + 4), (lv4i_p)(&s_sh[base + 4]), 0, 0);
    __builtin_amdgcn_global_load_async_to_lds_b128((gv4i_p)(gl),     (lv4i_p)(&l_sh[base]),     0, 0);
    __builtin_amdgcn_global_load_async_to_lds_b128((gv4i_p)(gl + 4), (lv4i_p)(&l_sh[base + 4]), 0, 0);
    wait_async_lds();
#else
    {
        float4 a = ((const float4*)gs)[0];
        float4 b = ((const float4*)gs)[1];
        float4 c = ((const float4*)gl)[0];
        float4 d = ((const float4*)gl)[1];
        ((float4*)&s_sh[base])[0] = a;
        ((float4*)&s_sh[base])[1] = b;
        ((float4*)&l_sh[base])[0] = c;
        ((float4*)&l_sh[base])[1] = d;
    }
#endif
    if (t < 5) hist[t] = 0;
    __syncthreads();

    // Pull my 8 elements into registers.
    float sv[EPT], lv[EPT];
#pragma unroll
    for (int j = 0; j < EPT; ++j) {
        sv[j] = s_sh[base + j];
        lv[j] = l_sh[base + j];
    }

    // ---- row min (for make_pos_by_subtraction) ----
    float mn = sv[0];
#pragma unroll
    for (int j = 1; j < EPT; ++j) mn = fminf(mn, sv[j]);
#pragma unroll
    for (int o = 16; o > 0; o >>= 1) mn = fminf(mn, __shfl_xor(mn, o, 32));
    if (lane == 0) red[wid] = mn;
    __syncthreads();
    float rmin = red[0];
#pragma unroll
    for (int i = 1; i < 8; ++i) rmin = fminf(rmin, red[i]);
    __syncthreads();
#pragma unroll
    for (int j = 0; j < EPT; ++j) sv[j] -= rmin;

    // ---- label histogram (labels are exactly {0,1,2,3,4}) -> exact top-10 ----
    {
        int c0 = 0, c1 = 0, c2 = 0, c3 = 0, c4 = 0;
#pragma unroll
        for (int j = 0; j < EPT; ++j) {
            float L = lv[j];
            c0 += (L == 0.0f);
            c1 += (L == 1.0f);
            c2 += (L == 2.0f);
            c3 += (L == 3.0f);
            c4 += (L == 4.0f);
        }
        atomicAdd(&hist[0], c0);
        atomicAdd(&hist[1], c1);
        atomicAdd(&hist[2], c2);
        atomicAdd(&hist[3], c3);
        atomicAdd(&hist[4], c4);
    }
    __syncthreads();

    // IDCG from histogram (computed redundantly by every thread; deterministic).
    float idcg = EPS_C;
    {
        int pos = 0;
        for (int v = 4; v >= 0; --v) {
            int c = hist[v];
            float pw = (float)(1 << v);  // exp2(label)
            for (int i = 0; i < c && pos < KTOP; ++i) {
                idcg += pw * fast_rcp(fast_log2((float)(pos + 2)));
                ++pos;
            }
        }
    }

    // ---- K iterations of smoothed argmax ----
    float prod[EPT];
#pragma unroll
    for (int j = 0; j < EPT; ++j) prod[j] = 1.0f;

    float dcg = EPS_C;
    for (int k = 0; k < KTOP; ++k) {
        // B = ALPHA * s * prod ; row max for stable softmax
        float e[EPT];
        float bmax = -INFINITY;
#pragma unroll
        for (int j = 0; j < EPT; ++j) {
            float b = ALPHA_C * sv[j] * prod[j];
            e[j] = b;
            bmax = fmaxf(bmax, b);
        }
#pragma unroll
        for (int o = 16; o > 0; o >>= 1) bmax = fmaxf(bmax, __shfl_xor(bmax, o, 32));
        if (lane == 0) red[wid] = bmax;
        __syncthreads();
        float m = red[0];
#pragma unroll
        for (int i = 1; i < 8; ++i) m = fmaxf(m, red[i]);
        __syncthreads();

        // exp, and the two row sums (Z and sum(label * e)) in one reduction
        float se = 0.0f, sl = 0.0f;
#pragma unroll
        for (int j = 0; j < EPT; ++j) {
            float ex = fast_exp2((e[j] - m) * LOG2E_C);
            e[j] = ex;
            se += ex;
            sl += lv[j] * ex;
        }
#pragma unroll
        for (int o = 16; o > 0; o >>= 1) {
            se += __shfl_xor(se, o, 32);
            sl += __shfl_xor(sl, o, 32);
        }
        if (lane == 0) red2[wid] = make_float2(se, sl);
        __syncthreads();
        float Z = red2[0].x, Ls = red2[0].y;
#pragma unroll
        for (int i = 1; i < 8; ++i) {
            Z += red2[i].x;
            Ls += red2[i].y;
        }
        __syncthreads();

        float invZ = fast_rcp(Z);
        float rel = Ls * invZ;  // sum(label * softmax)
#pragma unroll
        for (int j = 0; j < EPT; ++j) prod[j] *= (1.0f - e[j] * invZ - DELTA_C);

        dcg += fast_exp2(rel) * fast_rcp(fast_log2((float)(k + 2)));
    }

    if (t == 0) row_out[row] = 1.0f - dcg * fast_rcp(idcg);
}

// Deterministic single-block final reduction (fixed strided order + fixed tree).
__global__ __launch_bounds__(256) void final_sum_kernel(const float* __restrict__ in,
                                                        float* __restrict__ out, int n) {
    __shared__ float sm[256];
    float a = 0.0f;
    for (int i = threadIdx.x; i < n; i += 256) a += in[i];
    sm[threadIdx.x] = a;
    __syncthreads();
    for (int s = 128; s > 0; s >>= 1) {
        if (threadIdx.x < s) sm[threadIdx.x] += sm[threadIdx.x + s];
        __syncthreads();
    }
    if (threadIdx.x == 0) out[0] = sm[0];
}

extern "C" void kernel_launch(void* const* d_in, const int* in_sizes, int n_in,
                              void* d_out, int out_size, void* d_ws, size_t ws_size,
                              hipStream_t stream) {
    const float* s   = (const float*)d_in[0];
    const float* lab = (const float*)d_in[1];
    float* ws  = (float*)d_ws;
    float* out = (float*)d_out;

    const int total = in_sizes[0];
    const int bs = total / LL;  // 16384 rows

    ndcg_rows_kernel<<<bs, TPB, 0, stream>>>(s, lab, ws);
    final_sum_kernel<<<1, 256, 0, stream>>>(ws, out, bs);
}